// LevenshteinLoss_40003325395223
// MI455X (gfx1250) — compile-verified
//
#include <hip/hip_runtime.h>

// Levenshtein DP (2048 x 2048), anti-diagonal wavefront, row-blocked in time.
// One workgroup, 256 threads (8 wave32s), 8 columns per thread in VGPRs.
// Each step a thread computes R=4 rows of its stripe (32 cells) and exchanges
// 4 boundary values as one b128 LDS transaction -> 3x fewer barriers (767 steps).
// seq1 staged into LDS with CDNA5 async global->LDS loads (ASYNCcnt path).

#define TB     256            // threads per block (8 wave32s)
#define WCOLS  8              // columns owned per thread (TB*WCOLS == 2048)
#define RROWS  4              // rows per thread per step (barrier amortization)
#define NROWS  2048           // seq1 length (rows)
#define NB     512            // ceil((NROWS-1)/RROWS) row-blocks (rows 1..2047)
#define BIGV   (1 << 20)      // "+inf" for column-0 boundary folding

__global__ __launch_bounds__(TB) void lev_wavefront_kernel(
    const float* __restrict__ seq1,
    const float* __restrict__ seq2,
    float* __restrict__ out) {
  __shared__ float sSeq1[NROWS];                       // row tokens (8 KB)
  __shared__ __align__(16) int pub[2][TB * RROWS];     // parity-buffered boundaries (8 KB)

  const int tid = threadIdx.x;

  // ---- CDNA5 async stage: seq1 (8 KB) global -> LDS, tracked by ASYNCcnt ----
  {
    const uint32_t ldsBase = (uint32_t)(uintptr_t)(&sSeq1[0]); // flat low 32b == LDS offset
#pragma unroll
    for (int i = 0; i < NROWS / TB; ++i) {
      const uint32_t e       = (uint32_t)(tid + i * TB);
      const uint32_t ldsAddr = ldsBase + e * 4u;   // per-lane LDS byte address (VDST)
      const uint32_t goff    = e * 4u;             // per-lane global byte offset (VADDR, GVS)
      asm volatile("global_load_async_to_lds_b32 %0, %1, %2"
                   :
                   : "v"(ldsAddr), "v"(goff), "s"(seq1)
                   : "memory");
    }
  }

  // ---- per-thread seq2 stripe into VGPRs (2 x b128, overlaps the async copy) ----
  float s2[WCOLS];
  {
    const float4* p = (const float4*)(seq2 + tid * WCOLS);
    const float4 q0 = p[0];
    const float4 q1 = p[1];
    s2[0] = q0.x; s2[1] = q0.y; s2[2] = q0.z; s2[3] = q0.w;
    s2[4] = q1.x; s2[5] = q1.y; s2[6] = q1.z; s2[7] = q1.w;
  }

  // ---- DP state: row 0 is D[0][j] = j ----
  int v[WCOLS];
#pragma unroll
  for (int k = 0; k < WCOLS; ++k) v[k] = tid * WCOLS + k;
  // diag seed for the first row of the next block: D[r0-1][c0-1]
  int prevLast = (tid == 0) ? BIGV : (tid * WCOLS - 1);

  asm volatile("s_wait_asynccnt 0" ::: "memory");  // our async LDS writes done
  __syncthreads();                                 // all waves' writes visible

  // ---- wavefront over row-blocks: thread `tid` runs block b = t - tid ----
  const int STEPS = NB + TB - 1;  // 767
  for (int t = 1; t <= STEPS; ++t) {
    const int b = t - tid;
    if (b >= 1 && b <= NB) {
      // neighbor's right-column values for this block's rows (one ds_load_b128)
      const int4 linv = (tid == 0)
                            ? make_int4(BIGV, BIGV, BIGV, BIGV)
                            : *(const int4*)&pub[(t - 1) & 1][(tid - 1) * RROWS];
      const int lin[RROWS] = {linv.x, linv.y, linv.z, linv.w};
      int outv[RROWS] = {0, 0, 0, 0};
      const int dfirst = prevLast;
      prevLast = lin[RROWS - 1];
      const int r0 = (b - 1) * RROWS + 1;
#pragma unroll
      for (int rr = 0; rr < RROWS; ++rr) {
        const int r = r0 + rr;
        if (r <= NROWS - 1) {                 // only the last block is partial
          const float a = sSeq1[r];           // conflict-free rotating index
          int l = lin[rr];                    // D[r][c0-1]
          int d = (rr == 0) ? dfirst : lin[rr - 1];  // D[r-1][c0-1]
#pragma unroll
          for (int k = 0; k < WCOLS; ++k) {
            const int cost = (a == s2[k]) ? 0 : 1;
            const int nv   = min(min(v[k], l) + 1, d + cost);  // up+1,left+1,diag+cost
            d = v[k];
            l = nv;
            v[k] = nv;
          }
          outv[rr] = l;                       // D[r][c0+WCOLS-1]
        }
      }
      // publish this block's right-column values (one ds_store_b128)
      *(int4*)&pub[t & 1][tid * RROWS] =
          make_int4(outv[0], outv[1], outv[2], outv[3]);
    }
    __syncthreads();  // one split barrier per 4 anti-diagonal rows
  }

  if (tid == TB - 1) out[0] = (float)v[WCOLS - 1];  // D[2047][2047]
}

extern "C" void kernel_launch(void* const* d_in, const int* in_sizes, int n_in,
                              void* d_out, int out_size, void* d_ws, size_t ws_size,
                              hipStream_t stream) {
  (void)in_sizes; (void)n_in; (void)out_size; (void)d_ws; (void)ws_size;
  const float* seq1 = (const float*)d_in[0];
  const float* seq2 = (const float*)d_in[1];
  float* out = (float*)d_out;
  hipLaunchKernelGGL(lev_wavefront_kernel, dim3(1), dim3(TB), 0, stream,
                     seq1, seq2, out);
}